// MaskedGNN_22471268893325
// MI455X (gfx1250) — compile-verified
//
#include <hip/hip_runtime.h>
#include <hip/hip_bf16.h>

typedef __attribute__((ext_vector_type(2))) float v2f;
typedef __attribute__((ext_vector_type(8))) float v8f;

#define WIDTH 128

// ---------------- degree / normalization ----------------
__global__ void k_init_deg(float* dis, int n) {
    int i = blockIdx.x * blockDim.x + threadIdx.x;
    if (i < n) dis[i] = 1.0f;   // self-loop
}

__global__ void k_deg_accum(const int* dst, float* dis, int E) {
    int e = blockIdx.x * blockDim.x + threadIdx.x;
    if (e < E) atomicAdd(&dis[dst[e]], 1.0f);
}

__global__ void k_finish_dis(float* dis, int n) {
    int i = blockIdx.x * blockDim.x + threadIdx.x;
    if (i < n) dis[i] = rsqrtf(dis[i]);
}

// ---------------- encoder: t = x @ W_enc  (K = 3) ----------------
__global__ void k_encoder(const float* __restrict__ x, const float* __restrict__ We,
                          float* __restrict__ t, int n) {
    int i = blockIdx.x;            // node
    int j = threadIdx.x;           // feature 0..127
    if (i >= n) return;
    float x0 = x[i * 3 + 0], x1 = x[i * 3 + 1], x2 = x[i * 3 + 2];
    t[(size_t)i * WIDTH + j] = x0 * We[0 * WIDTH + j] + x1 * We[1 * WIDTH + j] + x2 * We[2 * WIDTH + j];
}

// ---------------- WMMA GEMM: t = relu(h) @ Wt   [n x 128] @ [128 x 128] ----------------
// One wave per 16x16 output tile. K-loop of 32 x V_WMMA_F32_16X16X4_F32.
__global__ void k_gemm_wmma(const float* __restrict__ h, const float* __restrict__ Wt,
                            float* __restrict__ t, int tilesM) {
    const int tilesN = WIDTH / 16;                       // 8
    int wave = blockIdx.x * (blockDim.x >> 5) + (threadIdx.x >> 5);
    if (wave >= tilesM * tilesN) return;                 // whole-wave exit: EXEC stays all-1s
    int tileM = wave / tilesN;
    int tileN = wave % tilesN;
    int lane  = threadIdx.x & 31;
    int lo    = lane & 15;
    int hi    = lane >> 4;                               // 0: lanes 0-15, 1: lanes 16-31

    int row0 = tileM * 16;
    int col0 = tileN * 16;

    const float* hrow = h + (size_t)(row0 + lo) * WIDTH; // A row for this lane (M = lo)
    int colB = col0 + lo;                                // B column for this lane (N = lo)

    v8f c = {};
    for (int k = 0; k < WIDTH; k += 4) {
        int kk = k + 2 * hi;                             // lanes 0-15: K=k,k+1 ; lanes 16-31: K=k+2,k+3
        v2f a = *(const v2f*)(hrow + kk);
        a.x = fmaxf(a.x, 0.0f);                          // fused ReLU on input activations
        a.y = fmaxf(a.y, 0.0f);
        v2f b;
        b.x = Wt[(size_t)kk * WIDTH + colB];
        b.y = Wt[(size_t)(kk + 1) * WIDTH + colB];
        c = __builtin_amdgcn_wmma_f32_16x16x4_f32(
                /*neg_a=*/false, a, /*neg_b=*/false, b,
                /*c_mod=*/(short)0, c, /*reuse_a=*/false, /*reuse_b=*/false);
    }

    // D layout: VGPR r -> M = r (lanes 0-15) / M = r+8 (lanes 16-31), N = lane%16
    float* trow = t + (size_t)(row0 + hi * 8) * WIDTH + col0 + lo;
    #pragma unroll
    for (int r = 0; r < 8; ++r) trow[(size_t)r * WIDTH] = c[r];
}

// ---------------- self-loop + bias init: acc = t * dis^2 + b ----------------
__global__ void k_selfloop(const float* __restrict__ t, const float* __restrict__ dis,
                           const float* __restrict__ b, float* __restrict__ acc, int n) {
    int idx = blockIdx.x * blockDim.x + threadIdx.x;
    if (idx >= n * WIDTH) return;
    int i = idx >> 7;
    int j = idx & (WIDTH - 1);
    float d = dis[i];
    acc[idx] = t[idx] * d * d + b[j];
}

// ---------------- edge scatter: acc[dst] += t[src] * (dis[src]*dis[dst]) ----------------
__global__ void k_scatter(const float* __restrict__ t, const int* __restrict__ src,
                          const int* __restrict__ dst, const float* __restrict__ dis,
                          float* __restrict__ acc, int E) {
    long long idx = (long long)blockIdx.x * blockDim.x + threadIdx.x;
    long long total = (long long)E * WIDTH;
    if (idx >= total) return;
    int e = (int)(idx >> 7);
    int j = (int)(idx & (WIDTH - 1));
    int s = src[e], d = dst[e];
    float norm = dis[s] * dis[d];
    atomicAdd(&acc[(size_t)d * WIDTH + j], t[(size_t)s * WIDTH + j] * norm);
}

// ---------------- decoder dot: tdec[i] = relu(h[i,:]) . Wd  (wave per node) ----------------
__global__ void k_dec_dot(const float* __restrict__ h, const float* __restrict__ Wd,
                          float* __restrict__ tdec, int n) {
    int gwave = blockIdx.x * (blockDim.x >> 5) + (threadIdx.x >> 5);
    if (gwave >= n) return;
    int lane = threadIdx.x & 31;
    const float4* hv = (const float4*)(h + (size_t)gwave * WIDTH);
    float4 v = hv[lane];
    float s = fmaxf(v.x, 0.0f) * Wd[lane * 4 + 0]
            + fmaxf(v.y, 0.0f) * Wd[lane * 4 + 1]
            + fmaxf(v.z, 0.0f) * Wd[lane * 4 + 2]
            + fmaxf(v.w, 0.0f) * Wd[lane * 4 + 3];
    #pragma unroll
    for (int off = 16; off > 0; off >>= 1) s += __shfl_down(s, off, 32);
    if (lane == 0) tdec[gwave] = s;
}

__global__ void k_dec_selfloop(const float* __restrict__ tdec, const float* __restrict__ dis,
                               const float* __restrict__ bdec, float* __restrict__ adec, int n) {
    int i = blockIdx.x * blockDim.x + threadIdx.x;
    if (i >= n) return;
    float d = dis[i];
    adec[i] = tdec[i] * d * d + bdec[0];
}

__global__ void k_dec_scatter(const float* __restrict__ tdec, const int* __restrict__ src,
                              const int* __restrict__ dst, const float* __restrict__ dis,
                              float* __restrict__ adec, int E) {
    int e = blockIdx.x * blockDim.x + threadIdx.x;
    if (e >= E) return;
    int s = src[e], d = dst[e];
    atomicAdd(&adec[d], tdec[s] * dis[s] * dis[d]);
}

__global__ void k_final(const float* __restrict__ adec, const float* __restrict__ mask,
                        float* __restrict__ out, int n) {
    int i = blockIdx.x * blockDim.x + threadIdx.x;
    if (i < n) out[i] = adec[i] * mask[i];
}

extern "C" void kernel_launch(void* const* d_in, const int* in_sizes, int n_in,
                              void* d_out, int out_size, void* d_ws, size_t ws_size,
                              hipStream_t stream) {
    const float* x     = (const float*)d_in[0];
    const float* mask  = (const float*)d_in[1];
    const int*   ei    = (const int*)d_in[2];
    const float* W_enc = (const float*)d_in[3];
    const float* b_enc = (const float*)d_in[4];
    const float* W_p   = (const float*)d_in[5];
    const float* b_p   = (const float*)d_in[6];
    const float* W_dec = (const float*)d_in[7];
    const float* b_dec = (const float*)d_in[8];
    float* out = (float*)d_out;

    const int N = in_sizes[0] / 3;      // 100000
    const int E = in_sizes[2] / 2;      // 1600000
    const int L = 4;
    const int* src = ei;
    const int* dst = ei + E;

    float* ws   = (float*)d_ws;
    float* dis  = ws;                   // [N]
    float* tdec = ws + N;               // [N]
    float* adec = ws + 2 * (size_t)N;   // [N]
    float* hbuf = ws + 3 * (size_t)N;                       // [N*128] aggregation/input buffer
    float* tbuf = hbuf + (size_t)N * WIDTH;                 // [N*128] GEMM output buffer

    const int B = 256;
    int bN  = (N + B - 1) / B;
    int bE  = (E + B - 1) / B;
    int bNW = (int)(((long long)N * WIDTH + B - 1) / B);
    int bEW = (int)(((long long)E * WIDTH + B - 1) / B);
    int tilesM     = N / 16;                                // 6250 (exact)
    int gemmBlocks = (tilesM * (WIDTH / 16) + 7) / 8;       // 8 waves of 32 per block
    int decBlocks  = (N * 32 + B - 1) / B;

    // 1) dis = rsqrt(1 + in-degree)
    k_init_deg<<<bN, B, 0, stream>>>(dis, N);
    k_deg_accum<<<bE, B, 0, stream>>>(dst, dis, E);
    k_finish_dis<<<bN, B, 0, stream>>>(dis, N);

    // 2) encoder GCN
    k_encoder<<<N, WIDTH, 0, stream>>>(x, W_enc, tbuf, N);
    k_selfloop<<<bNW, B, 0, stream>>>(tbuf, dis, b_enc, hbuf, N);
    k_scatter<<<bEW, B, 0, stream>>>(tbuf, src, dst, dis, hbuf, E);

    // 3) processor layers (ReLU fused into GEMM A-load)
    for (int l = 0; l < L; ++l) {
        k_gemm_wmma<<<gemmBlocks, B, 0, stream>>>(hbuf, W_p + (size_t)l * WIDTH * WIDTH, tbuf, tilesM);
        k_selfloop<<<bNW, B, 0, stream>>>(tbuf, dis, b_p + (size_t)l * WIDTH, hbuf, N);
        k_scatter<<<bEW, B, 0, stream>>>(tbuf, src, dst, dis, hbuf, E);
    }

    // 4) decoder GCN + mask
    k_dec_dot<<<decBlocks, B, 0, stream>>>(hbuf, W_dec, tdec, N);
    k_dec_selfloop<<<bN, B, 0, stream>>>(tdec, dis, b_dec, adec, N);
    k_dec_scatter<<<bE, B, 0, stream>>>(tdec, src, dst, dis, adec, E);
    k_final<<<bN, B, 0, stream>>>(adec, mask, out, N);
}